// BilinearAttention_85134841742040
// MI455X (gfx1250) — compile-verified
//
#include <hip/hip_runtime.h>
#include <hip/hip_bf16.h>

#define BB 8
#define SS 4096
#define TT 2048
#define CC 256
#define QQ 256
#define NEGF (-3.402823466e38f)
#define LDSP 260   // padded row stride (floats) for 256-wide LDS tiles (bank-conflict free)
#define ALDSP 20   // padded row stride for 16-wide attn tile
#define NTILES (SS / 16)

typedef float v2f __attribute__((ext_vector_type(2)));
typedef float v8f __attribute__((ext_vector_type(8)));

static __device__ __forceinline__ v8f wmma_f32_k4(v2f a, v2f b, v8f c) {
  // D(16x16,f32) = A(16x4,f32) * B(4x16,f32) + C
  return __builtin_amdgcn_wmma_f32_16x16x4_f32(false, a, false, b, (short)0, c,
                                               false, false);
}

// ---------------- async global->LDS staging (CDNA5 GLOBAL_LOAD_ASYNC_TO_LDS) ----
#if defined(__gfx1250__) && __has_builtin(__builtin_amdgcn_global_load_async_to_lds_b128)
#define USE_ASYNC_LDS 1
typedef int v4i_vs __attribute__((__vector_size__(4 * sizeof(int))));
typedef __attribute__((address_space(1))) v4i_vs as1_v4i;
typedef __attribute__((address_space(3))) v4i_vs as3_v4i;
static __device__ __forceinline__ void cp16_async(const float* g, float* l) {
  __builtin_amdgcn_global_load_async_to_lds_b128((as1_v4i*)g, (as3_v4i*)l, 0, 0);
}
#if __has_builtin(__builtin_amdgcn_s_wait_asynccnt)
#define WAIT_ASYNC(n) __builtin_amdgcn_s_wait_asynccnt(n)
#else
#define WAIT_ASYNC(n) asm volatile("s_wait_asynccnt %0" ::"i"(n) : "memory")
#endif
#else
#define WAIT_ASYNC(n)
#endif

// Stage one 16 x 256 fp32 tile (rows stride `gstride` in global, LDSP in LDS).
// Async mode: issues 16*64/nthreads ASYNC copy instructions per wave and returns
// immediately (caller waits). Sync mode: plain load+ds_store.
static __device__ __forceinline__ void stage_tile(const float* __restrict__ g,
                                                  int gstride, float* l, int tid,
                                                  int nthreads) {
  for (int i = tid; i < 16 * 64; i += nthreads) {
    const int r = i >> 6, c4 = (i & 63) << 2;
#ifdef USE_ASYNC_LDS
    cp16_async(g + (size_t)r * gstride + c4, l + r * LDSP + c4);
#else
    const float4 v = *(const float4*)(g + (size_t)r * gstride + c4);
    float* d = l + r * LDSP + c4;
    d[0] = v.x; d[1] = v.y; d[2] = v.z; d[3] = v.w;
#endif
  }
}

// ---------------------------------------------------------------------------
// Kernel 1: P[r, q] = (context[r, :] @ W[:, q]) * (1/sqrt(Q)),  r = b*SS + s
// ---------------------------------------------------------------------------
__global__ __launch_bounds__(32) void proj_kernel(const float* __restrict__ ctx,
                                                  const float* __restrict__ W,
                                                  float* __restrict__ P) {
  __shared__ float tl[16 * LDSP];
  const int q0 = blockIdx.x * 16;
  const int r0 = blockIdx.y * 16;
  const int lane = threadIdx.x;
  const int hi = lane >> 4, lm = lane & 15;

  stage_tile(ctx + (size_t)r0 * CC, CC, tl, lane, 32);
  WAIT_ASYNC(0);
  __syncthreads();

  v8f acc = {};
#pragma unroll 4
  for (int k = 0; k < CC; k += 4) {
    v2f a, b;
    a.x = tl[lm * LDSP + k + 2 * hi];
    a.y = tl[lm * LDSP + k + 2 * hi + 1];
    b.x = W[(size_t)(k + 2 * hi) * QQ + q0 + lm];
    b.y = W[(size_t)(k + 1 + 2 * hi) * QQ + q0 + lm];
    acc = wmma_f32_k4(a, b, acc);
  }
#pragma unroll
  for (int v = 0; v < 8; ++v)
    P[(size_t)(r0 + v + 8 * hi) * QQ + q0 + lm] = acc[v] * 0.0625f;  // 1/sqrt(256)
}

// ---------------------------------------------------------------------------
// Kernel 2: fused masked-softmax attention + composition for one (b, t-tile).
// Block = 64 threads = 2 waves; double-buffered async LDS staging of the
// P-tile (both passes) and context-tile (pass B) overlaps DMA with WMMA.
// ---------------------------------------------------------------------------
__global__ __launch_bounds__(64) void attn_kernel(const float* __restrict__ ctx,
                                                  const float* __restrict__ qry,
                                                  const unsigned char* __restrict__ mask,
                                                  const float* __restrict__ P,
                                                  float* __restrict__ attn_out,
                                                  float* __restrict__ comp_out) {
  __shared__ float qlds[16 * LDSP];
  __shared__ float pbuf[2][16 * LDSP];
  __shared__ float cbuf[2][16 * LDSP];
  __shared__ float alds[16 * ALDSP];

  const int b = blockIdx.y;
  const int t0 = blockIdx.x * 16;
  const int tid = threadIdx.x;
  const int wid = tid >> 5;
  const int lane = tid & 31;
  const int hi = lane >> 4, lm = lane & 15;

  const float* Pb = P + (size_t)b * SS * QQ;
  const float* cb = ctx + (size_t)b * SS * CC;
  const unsigned char* mb = mask + (size_t)b * SS;

  // all-masked flag for batch b (reference zeroes attn for such batches)
  int am = 1;
  for (int i = lane; i < SS; i += 32) am &= (mb[i] != 0) ? 1 : 0;
#pragma unroll
  for (int d = 1; d < 32; d <<= 1) am &= __shfl_xor(am, d, 32);

  // stage query tile [16 x QQ] once; completion covered by first WAIT+barrier
  {
    const float* qt = qry + ((size_t)t0 * BB + b) * QQ;
    for (int i = tid; i < 16 * 64; i += 64) {
      const int r = i >> 6, c4 = (i & 63) << 2;
#ifdef USE_ASYNC_LDS
      cp16_async(qt + (size_t)r * BB * QQ + c4, &qlds[r * LDSP + c4]);
#else
      const float4 v = *(const float4*)(qt + (size_t)r * BB * QQ + c4);
      float* d = &qlds[r * LDSP + c4];
      d[0] = v.x; d[1] = v.y; d[2] = v.z; d[3] = v.w;
#endif
    }
  }

  float mrow[8], srow[8];
#pragma unroll
  for (int v = 0; v < 8; ++v) { mrow[v] = NEGF; srow[v] = 0.0f; }

  // ------------------------- pass A: online row max / sumexp -----------------
  stage_tile(Pb, QQ, pbuf[0], tid, 64);
  for (int it = 0; it < NTILES; ++it) {
    const int s0 = it * 16;
    __syncthreads();  // previous compute done before overwriting next buffer
    if (it + 1 < NTILES) {
      stage_tile(Pb + (size_t)(s0 + 16) * QQ, QQ, pbuf[(it + 1) & 1], tid, 64);
      WAIT_ASYNC(16);  // 16 newer copies in flight -> current tile resident
    } else {
      WAIT_ASYNC(0);
    }
    __syncthreads();
    const float* pl = pbuf[it & 1];

    v8f sc = {};
#pragma unroll 4
    for (int k = 0; k < QQ; k += 4) {
      v2f a, bb;
      a.x = qlds[lm * LDSP + k + 2 * hi];
      a.y = qlds[lm * LDSP + k + 2 * hi + 1];
      bb.x = pl[lm * LDSP + k + 2 * hi];  // B[k][n] = P[s0+n][k]
      bb.y = pl[lm * LDSP + k + 2 * hi + 1];
      sc = wmma_f32_k4(a, bb, sc);
    }

    const bool mk = (mb[s0 + lm] != 0);
#pragma unroll
    for (int v = 0; v < 8; ++v) {
      const float x = mk ? NEGF : sc[v];
      float t = x;
#pragma unroll
      for (int d = 1; d < 16; d <<= 1) t = fmaxf(t, __shfl_xor(t, d, 16));
      const float mnew = fmaxf(mrow[v], t);
      float e = __expf(x - mnew);
#pragma unroll
      for (int d = 1; d < 16; d <<= 1) e += __shfl_xor(e, d, 16);
      srow[v] = srow[v] * __expf(mrow[v] - mnew) + e;
      mrow[v] = mnew;
    }
  }

  float inv[8];
#pragma unroll
  for (int v = 0; v < 8; ++v) inv[v] = 1.0f / srow[v];

  const v8f zacc = {};
  v8f acc[8];
#pragma unroll
  for (int n = 0; n < 8; ++n) acc[n] = zacc;
  const int n0base = wid * 128;

  // ------------------------- pass B: attn out + composition ------------------
  stage_tile(Pb, QQ, pbuf[0], tid, 64);
  stage_tile(cb, CC, cbuf[0], tid, 64);
  for (int it = 0; it < NTILES; ++it) {
    const int s0 = it * 16;
    __syncthreads();
    if (it + 1 < NTILES) {
      stage_tile(Pb + (size_t)(s0 + 16) * QQ, QQ, pbuf[(it + 1) & 1], tid, 64);
      stage_tile(cb + (size_t)(s0 + 16) * CC, CC, cbuf[(it + 1) & 1], tid, 64);
      WAIT_ASYNC(32);  // 32 newer copies in flight -> current tiles resident
    } else {
      WAIT_ASYNC(0);
    }
    __syncthreads();
    const float* pl = pbuf[it & 1];
    const float* cl = cbuf[it & 1];

    v8f sc = {};
#pragma unroll 4
    for (int k = 0; k < QQ; k += 4) {
      v2f a, bb;
      a.x = qlds[lm * LDSP + k + 2 * hi];
      a.y = qlds[lm * LDSP + k + 2 * hi + 1];
      bb.x = pl[lm * LDSP + k + 2 * hi];
      bb.y = pl[lm * LDSP + k + 2 * hi + 1];
      sc = wmma_f32_k4(a, bb, sc);
    }

    const bool mk = (mb[s0 + lm] != 0);
    float av[8];
#pragma unroll
    for (int v = 0; v < 8; ++v) {
      const float x = mk ? NEGF : sc[v];
      av[v] = am ? 0.0f : __expf(x - mrow[v]) * inv[v];
    }
    if (wid == 0) {
#pragma unroll
      for (int v = 0; v < 8; ++v) {
        const int row = v + 8 * hi;
        attn_out[((size_t)(t0 + row) * BB + b) * SS + s0 + lm] = av[v];
        alds[row * ALDSP + lm] = av[v];
      }
    }
    __syncthreads();

    // composition: acc[t, c] += attn_tile(16x16) @ ctx_tile(16 x 128-half)
#pragma unroll
    for (int nt = 0; nt < 8; ++nt) {
      const int n0 = n0base + nt * 16;
#pragma unroll
      for (int kk = 0; kk < 16; kk += 4) {
        v2f a, bb;
        a.x = alds[lm * ALDSP + kk + 2 * hi];
        a.y = alds[lm * ALDSP + kk + 2 * hi + 1];
        bb.x = cl[(kk + 2 * hi) * LDSP + n0 + lm];
        bb.y = cl[(kk + 1 + 2 * hi) * LDSP + n0 + lm];
        acc[nt] = wmma_f32_k4(a, bb, acc[nt]);
      }
    }
  }

#pragma unroll
  for (int nt = 0; nt < 8; ++nt)
#pragma unroll
    for (int v = 0; v < 8; ++v)
      comp_out[((size_t)(t0 + v + 8 * hi) * BB + b) * CC + n0base + nt * 16 + lm] =
          acc[nt][v];
}

// ---------------------------------------------------------------------------
extern "C" void kernel_launch(void* const* d_in, const int* in_sizes, int n_in,
                              void* d_out, int out_size, void* d_ws, size_t ws_size,
                              hipStream_t stream) {
  const float* ctx = (const float*)d_in[0];                   // [B, S, C]
  const float* qry = (const float*)d_in[1];                   // [T, B, Q]
  const unsigned char* mask = (const unsigned char*)d_in[2];  // [B, S] bool
  const float* W = (const float*)d_in[3];                     // [C, Q]

  float* P = (float*)d_ws;                            // [B, S, Q] = 33.5 MB scratch
  float* attn_out = (float*)d_out;                    // [T, B, S]
  float* comp_out = attn_out + (size_t)TT * BB * SS;  // [T, B, C]

  dim3 g1(QQ / 16, (BB * SS) / 16);
  proj_kernel<<<g1, 32, 0, stream>>>(ctx, W, P);

  dim3 g2(TT / 16, BB);
  attn_kernel<<<g2, 64, 0, stream>>>(ctx, qry, mask, P, attn_out, comp_out);
}